// PairwiseAtomicEmbedding_89618787598480
// MI455X (gfx1250) — compile-verified
//
#include <hip/hip_runtime.h>

// ---------------------------------------------------------------------------
// PairwiseAtomicEmbedding for MI455X (gfx1250, wave32)
//   out[e, 0:400]   = exp(-(( |s_i - d_j + eps| - mu_k)/1.25)^2), p=i*5+j, k=0..15
//   out[e, 400:416] = cos/sin(d * 10^{-q/2})
// Pairwise distances via V_WMMA_F32_16X16X4_F32:
//   A_i=(sx',sy',sz',|s'|^2), B_j=(-2dx,-2dy,-2dz,1), C_ij=|d_j|^2  =>  D=|s'-d|^2
// 3 edges (15 src atoms x 15 dst atoms) per 16x16 WMMA tile per wave.
// Memory-bound: 1.065 GB of output @ 23.3 TB/s ~= 46 us floor; compute kept
// off the critical path (hardware v_sqrt/v_exp/v_cos/v_sin, dual-issue VALU).
// ---------------------------------------------------------------------------

typedef __attribute__((ext_vector_type(2))) float v2f;
typedef __attribute__((ext_vector_type(8))) float v8f;

#define EPSC 1e-8f

__constant__ float c_freq[8] = {
    1.0f, 0.31622776601683794f, 0.1f, 0.031622776601683794f,
    0.01f, 0.0031622776601683794f, 0.001f, 0.00031622776601683794f};

// ---- Kernel 1: bb (n,4,3) -> bb5 (n,5,3) with ideal virtual CB appended ----
__global__ void build_bb5_kernel(const float* __restrict__ bb,
                                 float* __restrict__ bb5, int n) {
  int i = blockIdx.x * blockDim.x + threadIdx.x;
  if (i >= n) return;
  const float* p = bb + (size_t)i * 12;
  float Nx = p[0], Ny = p[1], Nz = p[2];
  float Ax = p[3], Ay = p[4], Az = p[5];
  float Cx = p[6], Cy = p[7], Cz = p[8];
  float Ox = p[9], Oy = p[10], Oz = p[11];
  float bx = Ax - Nx, by = Ay - Ny, bz = Az - Nz;   // b = CA - N
  float cx = Cx - Ax, cy = Cy - Ay, cz = Cz - Az;   // c = C - CA
  float ax = by * cz - bz * cy;                     // a = cross(b, c)
  float ay = bz * cx - bx * cz;
  float az = bx * cy - by * cx;
  float vx = -0.58273431f * ax + 0.56802827f * bx - 0.54067466f * cx + Ax;
  float vy = -0.58273431f * ay + 0.56802827f * by - 0.54067466f * cy + Ay;
  float vz = -0.58273431f * az + 0.56802827f * bz - 0.54067466f * cz + Az;
  float* q = bb5 + (size_t)i * 15;
  q[0] = Nx; q[1] = Ny; q[2] = Nz;
  q[3] = Ax; q[4] = Ay; q[5] = Az;
  q[6] = Cx; q[7] = Cy; q[8] = Cz;
  q[9] = Ox; q[10] = Oy; q[11] = Oz;
  q[12] = vx; q[13] = vy; q[14] = vz;
}

// ---- Kernel 2: 8 waves/block, 3 edges/wave (24 edges/block) ----
__global__ __launch_bounds__(256) void edge_embed_kernel(
    const float* __restrict__ bb5,
    const int* __restrict__ e_dst,   // edge_index[0]
    const int* __restrict__ e_src,   // edge_index[1]
    float* __restrict__ out,
    long ne) {
  __shared__ float s_dist[8 * 256];          // 16x16 sqrt(D) tile per wave

  const int lane = threadIdx.x & 31;
  const int wave = threadIdx.x >> 5;
  const int r    = lane & 15;                // row (A) / col (B) in tile
  const int hi   = lane >> 4;                // 0: K=0,1   1: K=2,3
  const long e0  = (long)blockIdx.x * 24 + (long)wave * 3;

  const int et = r / 5;                      // edge within the wave's triple
  const int at = r % 5;                      // atom index 0..4

  // ---------- A fragment: src atoms (+eps folded in), rows ----------
  long eA = e0 + et;
  bool vA = (r < 15) && (eA < ne);
  long eAc = vA ? eA : 0;
  int sN = e_src[eAc];
  const float* sp = bb5 + (size_t)sN * 15 + at * 3;
  float sx = vA ? sp[0] + EPSC : 0.0f;
  float sy = vA ? sp[1] + EPSC : 0.0f;
  float sz = vA ? sp[2] + EPSC : 0.0f;
  float pS = hi ? sz * sz : sx * sx + sy * sy;
  pS += __shfl_xor(pS, 16, 32);              // |s'|^2 (both half-waves)
  v2f afrag;
  afrag.x = hi ? sz : sx;                    // K=2 : K=0
  afrag.y = hi ? pS : sy;                    // K=3 : K=1

  // ---------- B fragment: dst atoms, cols ----------
  long eB = e0 + et;
  bool vB = (r < 15) && (eB < ne);
  long eBc = vB ? eB : 0;
  int dN = e_dst[eBc];
  const float* dp = bb5 + (size_t)dN * 15 + at * 3;
  float dx = vB ? dp[0] : 0.0f;
  float dy = vB ? dp[1] : 0.0f;
  float dz = vB ? dp[2] : 0.0f;
  float pD = hi ? dz * dz : dx * dx + dy * dy;
  pD += __shfl_xor(pD, 16, 32);              // |d_j|^2 for col j = lane&15
  v2f bfrag;
  bfrag.x = hi ? -2.0f * dz : -2.0f * dx;    // K=2 : K=0
  bfrag.y = hi ? 1.0f       : -2.0f * dy;    // K=3 : K=1

  v8f cfrag;
#pragma unroll
  for (int v = 0; v < 8; ++v) cfrag[v] = pD; // C_ij = |d_j|^2 (col-constant)

  // D_ij = |s' - d|^2   (EXEC all-1s: all guards above are selects)
  v8f dmat = __builtin_amdgcn_wmma_f32_16x16x4_f32(
      false, afrag, false, bfrag, (short)0, cfrag, false, false);

  // scatter sqrt(D) to this wave's LDS tile: lane holds col r, rows v+8*hi
  // raw v_sqrt_f32 (no IEEE fixup chain) -- feeds a Gaussian, precision ample
  float* ld = s_dist + wave * 256;
#pragma unroll
  for (int v = 0; v < 8; ++v) {
    int M = v + 8 * hi;
    ld[M * 16 + r] = __builtin_amdgcn_sqrtf(fmaxf(dmat[v], 0.0f));
  }
  __syncthreads();

  // ---------- expand to 416 channels per edge, coalesced stores ----------
#pragma unroll
  for (int a = 0; a < 3; ++a) {
    long e = e0 + a;
    bool valid = e < ne;
    long ec = valid ? e : 0;
    float posd = (float)(e_dst[ec] - e_src[ec]);
    float* op = out + (size_t)ec * 416;
#pragma unroll
    for (int t = 0; t < 13; ++t) {           // 13*32 = 416 channels exactly
      int c = lane + 32 * t;
      float val;
      if (c < 400) {
        int p = c >> 4;                      // pair 0..24
        int i = p / 5, j = p % 5;
        float dist = ld[(5 * a + i) * 16 + (5 * a + j)];
        float u = (dist - (float)(c & 15) * (20.0f / 15.0f)) * 0.8f; // /1.25
        val = __expf(-u * u);                // v_exp_f32
      } else {
        int k = c - 400;
        float ang = posd * c_freq[k & 7];
        val = (k < 8) ? __cosf(ang) : __sinf(ang);
      }
      if (valid) op[c] = val;
    }
  }
}

extern "C" void kernel_launch(void* const* d_in, const int* in_sizes, int n_in,
                              void* d_out, int out_size, void* d_ws, size_t ws_size,
                              hipStream_t stream) {
  const float* bb = (const float*)d_in[0];   // (n,4,3) f32
  const int* ei   = (const int*)d_in[1];     // (2, ne) int
  int nn  = in_sizes[0] / 12;
  long ne = (long)(in_sizes[1] / 2);

  float* bb5 = (float*)d_ws;                 // (n,5,3) f32 = 60 B/node

  build_bb5_kernel<<<(nn + 255) / 256, 256, 0, stream>>>(bb, bb5, nn);

  long nblk = (ne + 23) / 24;                // 24 edges per 256-thread block
  edge_embed_kernel<<<(unsigned)nblk, 256, 0, stream>>>(
      bb5, ei, ei + ne, (float*)d_out, ne);
}